// MultiInputSTGACN_7447473291366
// MI455X (gfx1250) — compile-verified
//
#include <hip/hip_runtime.h>
#include <math.h>

// ---------------------------------------------------------------------------
// MultiInputSTGACN for MI455X (gfx1250, wave32).
//
// Roofline: head w1 [1024 x 262144] f32 = 1.07 GB read once -> ~46us at
// 23.3 TB/s dominates; implemented as a streaming float4 GEMV.
// Matrix-shaped work (K=5 convs via im2col, GATv2 projections) uses
// V_WMMA_F32_16X16X4_F32 with pre-padded/pre-packed operands so the inner
// loop is branch-free: one b64 A load + one b64 B load + one WMMA per step.
// ---------------------------------------------------------------------------

typedef float v2f __attribute__((ext_vector_type(2)));
typedef float v8f __attribute__((ext_vector_type(8)));

#define N_NODES 2048

// float atomic max via monotone bit mapping (valid for mixed signs with -inf init)
__device__ inline void atomicMaxF(float* addr, float v) {
    if (v >= 0.0f) {
        atomicMax((int*)addr, __float_as_int(v));
    } else {
        atomicMin((unsigned int*)addr, (unsigned int)__float_as_int(v));
    }
}

// ---------------------------------------------------------------------------
// f32 WMMA GEMM on pre-padded operands:
//   D[M, 2048] = act(A[M,Kpad] * B + bias[m]),  Kpad % 4 == 0 (zero padded)
//   A:  row-major [M, Kpad]
//   B:  pair-major packed: element (k,n) at B[(k>>1)*2*N + 2*n + (k&1)]
//       -> a lane's K-pair {k,k+1} for a given n is one contiguous v2f.
//   D element (m,n) at D[m*lddm + n*lddn]
// One wave per 16x16 D tile, 4 waves/block tiling N. No branches in loop,
// EXEC stays all-ones (WMMA requirement).
// Fragment layouts per CDNA5 ISA 7.12.2:
//   A 16x4 : lane l (m=l&15), half=l>>4 selects K-pair {k0,k0+1}/{k0+2,k0+3}
//   C/D    : vgpr r, lane l: M = r + 8*half, N = l&15
//   B 4x16 : mirror of A (n=l&15, half selects K-pair)
// ---------------------------------------------------------------------------
__global__ void wmma_gemm_f32(const float* __restrict__ A,
                              const float* __restrict__ B,
                              const float* __restrict__ bias,
                              float* __restrict__ D,
                              int Kpad, int lddm, int lddn, int do_relu)
{
    const int lane   = threadIdx.x & 31;
    const int waveid = threadIdx.x >> 5;
    const int n0   = (blockIdx.x * 4 + waveid) * 16;
    const int m0   = blockIdx.y * 16;
    const int half = lane >> 4;
    const int mr   = lane & 15;       // m for A-frag; n for B-frag / D store

    v8f acc = {};
    const float* Ap = A + (size_t)(m0 + mr) * Kpad + 2 * half;          // 8B aligned
    const float* Bp = B + (size_t)half * (2 * N_NODES) + 2 * (n0 + mr); // 8B aligned

    const int steps = Kpad >> 2;
    #pragma unroll 2
    for (int t = 0; t < steps; ++t) {
        const v2f a = *(const v2f*)Ap;
        const v2f b = *(const v2f*)Bp;
        Ap += 4;
        Bp += 4 * N_NODES;
        acc = __builtin_amdgcn_wmma_f32_16x16x4_f32(
                  false, a, false, b, (short)0, acc, false, false);
    }

    #pragma unroll
    for (int r = 0; r < 8; ++r) {
        const int m = m0 + r + 8 * half;
        float v = acc[r] + bias[m];
        if (do_relu) v = v > 0.0f ? v : 0.0f;
        D[(size_t)m * lddm + (size_t)(n0 + mr) * lddn] = v;
    }
}

// Zero-pad weight [M,K] row-major -> [M,Kpad]
__global__ void pack_w(const float* __restrict__ Wsrc, float* __restrict__ Wp,
                       int M, int K, int Kpad)
{
    const int idx = blockIdx.x * blockDim.x + threadIdx.x;
    if (idx >= M * Kpad) return;
    const int m = idx / Kpad;
    const int k = idx - m * Kpad;
    Wp[idx] = (k < K) ? Wsrc[m * K + k] : 0.0f;
}

// im2col for 'same' k=5 conv, written in pair-major packed form:
//   value(k=c*5+j, n) = X[c*N + n+j-2] (zero pad), dest (k>>1)*2N + 2n + (k&1)
__global__ void pack_im2col5(const float* __restrict__ X, float* __restrict__ Bp,
                             int Cin, int Kpad)
{
    const int idx = blockIdx.x * blockDim.x + threadIdx.x;
    if (idx >= Kpad * N_NODES) return;
    const int k = idx / N_NODES;
    const int n = idx - k * N_NODES;
    float v = 0.0f;
    if (k < Cin * 5) {
        const int c = k / 5;
        const int j = k - c * 5;
        const int t = n + j - 2;
        if (t >= 0 && t < N_NODES) v = X[c * N_NODES + t];
    }
    Bp[(size_t)(k >> 1) * (2 * N_NODES) + 2 * n + (k & 1)] = v;
}

// Raw-view projection operand, pair-major packed: value(k,n) = X[n*Cin + k]
__global__ void pack_rawview(const float* __restrict__ X, float* __restrict__ Bp,
                             int Cin, int Kpad)
{
    const int idx = blockIdx.x * blockDim.x + threadIdx.x;
    if (idx >= Kpad * N_NODES) return;
    const int k = idx / N_NODES;
    const int n = idx - k * N_NODES;
    const float v = (k < Cin) ? X[(size_t)n * Cin + k] : 0.0f;
    Bp[(size_t)(k >> 1) * (2 * N_NODES) + 2 * n + (k & 1)] = v;
}

__global__ void gat_init(float* __restrict__ m, float* __restrict__ den,
                         float* __restrict__ gat, int total)
{
    const int idx = blockIdx.x * blockDim.x + threadIdx.x;
    if (idx < N_NODES) { m[idx] = -__builtin_huge_valf(); den[idx] = 0.0f; }
    if (idx < total) gat[idx] = 0.0f;
}

// Pass 1: attention logit per (edge | self-loop) + segment max into mmax[d]
__global__ void edge_logits(const float* __restrict__ xl, const float* __restrict__ xr,
                            const float* __restrict__ att,
                            const int* __restrict__ src, const int* __restrict__ dst,
                            float* __restrict__ ebuf, float* __restrict__ mmax,
                            int E, int Ch)
{
    const int idx = blockIdx.x * blockDim.x + threadIdx.x;
    if (idx >= E + N_NODES) return;
    int s, d;
    if (idx < E) { s = src[idx]; d = dst[idx]; } else { s = d = idx - E; }
    const float4* pl = (const float4*)(xl + (size_t)s * Ch);
    const float4* pr = (const float4*)(xr + (size_t)d * Ch);
    const float4* pa = (const float4*)att;
    float e = 0.0f;
    const int C4 = Ch >> 2;
    for (int h = 0; h < C4; ++h) {
        const float4 a = pl[h], b = pr[h], t = pa[h];
        float z0 = a.x + b.x; z0 = z0 > 0.0f ? z0 : 0.2f * z0;
        float z1 = a.y + b.y; z1 = z1 > 0.0f ? z1 : 0.2f * z1;
        float z2 = a.z + b.z; z2 = z2 > 0.0f ? z2 : 0.2f * z2;
        float z3 = a.w + b.w; z3 = z3 > 0.0f ? z3 : 0.2f * z3;
        e += z0 * t.x + z1 * t.y + z2 * t.z + z3 * t.w;
    }
    ebuf[idx] = e;
    atomicMaxF(mmax + d, e);
}

// Pass 2: exp(e - max) + segment sum into den[d]
__global__ void edge_exp(float* __restrict__ ebuf, const float* __restrict__ mmax,
                         float* __restrict__ den, const int* __restrict__ dst, int E)
{
    const int idx = blockIdx.x * blockDim.x + threadIdx.x;
    if (idx >= E + N_NODES) return;
    const int d = (idx < E) ? dst[idx] : idx - E;
    const float ex = expf(ebuf[idx] - mmax[d]);
    ebuf[idx] = ex;
    atomicAdd(den + d, ex);
}

// Pass 3: alpha-weighted scatter of source features into gat[d,:]
__global__ void edge_aggr(const float* __restrict__ ebuf, const float* __restrict__ den,
                          const float* __restrict__ xl,
                          const int* __restrict__ src, const int* __restrict__ dst,
                          float* __restrict__ gat, int E, int Ch)
{
    const int idx = blockIdx.x * blockDim.x + threadIdx.x;
    if (idx >= E + N_NODES) return;
    int s, d;
    if (idx < E) { s = src[idx]; d = dst[idx]; } else { s = d = idx - E; }
    const float alpha = ebuf[idx] / den[d];
    const float4* pl = (const float4*)(xl + (size_t)s * Ch);
    float* pg = gat + (size_t)d * Ch;
    const int C4 = Ch >> 2;
    for (int h = 0; h < C4; ++h) {
        const float4 a = pl[h];
        atomicAdd(pg + 4 * h + 0, alpha * a.x);
        atomicAdd(pg + 4 * h + 1, alpha * a.y);
        atomicAdd(pg + 4 * h + 2, alpha * a.z);
        atomicAdd(pg + 4 * h + 3, alpha * a.w);
    }
}

// h1[n*Ch+c] = relu(BN(gat + gbias)),  c = idx % Ch   ([N,Ch] layout)
__global__ void bn_relu_nodefirst(const float* __restrict__ in,
                                  const float* __restrict__ gbias,
                                  const float* __restrict__ g, const float* __restrict__ bb,
                                  const float* __restrict__ mm, const float* __restrict__ vv,
                                  float* __restrict__ out, int total, int Ch)
{
    const int idx = blockIdx.x * blockDim.x + threadIdx.x;
    if (idx >= total) return;
    const int c = idx % Ch;
    float x = in[idx] + gbias[c];
    x = (x - mm[c]) * rsqrtf(vv[c] + 1e-5f) * g[c] + bb[c];
    out[idx] = x > 0.0f ? x : 0.0f;
}

// out[c*N+n] = res + relu(BN(tc)),  c = idx / N   ([Ch,N] layout)
__global__ void resbn_add(const float* __restrict__ tc, const float* __restrict__ res,
                          const float* __restrict__ g, const float* __restrict__ bb,
                          const float* __restrict__ mm, const float* __restrict__ vv,
                          float* __restrict__ out, int total)
{
    const int idx = blockIdx.x * blockDim.x + threadIdx.x;
    if (idx >= total) return;
    const int c = idx / N_NODES;
    float x = (tc[idx] - mm[c]) * rsqrtf(vv[c] + 1e-5f) * g[c] + bb[c];
    x = x > 0.0f ? x : 0.0f;
    out[idx] = res[idx] + x;
}

// Streaming GEMV: y[j] = act(dot(W[j,:], x) + bias[j]); one block per row,
// float4 loads; mode: 0=none, 1=relu, 2=sigmoid. Saturates HBM on w1 (1.07GB).
__global__ void gemv_act(const float* __restrict__ W, const float* __restrict__ x,
                         const float* __restrict__ bias, float* __restrict__ y,
                         int L, int mode)
{
    __shared__ float red[256];
    const int j = blockIdx.x;
    const float4* Wr = (const float4*)(W + (size_t)j * L);
    const float4* xv = (const float4*)x;
    const int L4 = L >> 2;
    float s = 0.0f;
    for (int i = threadIdx.x; i < L4; i += blockDim.x) {
        const float4 w = Wr[i];
        const float4 v = xv[i];
        s += w.x * v.x + w.y * v.y + w.z * v.z + w.w * v.w;
    }
    red[threadIdx.x] = s;
    __syncthreads();
    for (int off = blockDim.x >> 1; off > 0; off >>= 1) {
        if (threadIdx.x < off) red[threadIdx.x] += red[threadIdx.x + off];
        __syncthreads();
    }
    if (threadIdx.x == 0) {
        float t = red[0] + bias[j];
        if (mode == 1)      t = t > 0.0f ? t : 0.0f;
        else if (mode == 2) t = 1.0f / (1.0f + expf(-t));
        y[j] = t;
    }
}

// ---------------------------------------------------------------------------
// Host orchestration. Input order (setup_inputs dict / literal order):
//  [0] data0, [1] data1, [2] edge_index0, [3] edge_index1,
//  [4..143]  stream_params: 2 streams x 5 blocks x 14 leaves
//            (Wskip,bskip,Wl,bl,Wr,br,att,gbias,Wt,bt,g1,b1,m1,v1)
//  [144..149] head: w1,b1,w2,b2,w3,b3
// Workspace use: ~8.24 MB of f32 scratch.
// ---------------------------------------------------------------------------
extern "C" void kernel_launch(void* const* d_in, const int* in_sizes, int n_in,
                              void* d_out, int out_size, void* d_ws, size_t ws_size,
                              hipStream_t stream)
{
    (void)n_in; (void)out_size; (void)ws_size;
    const int N = N_NODES;
    const int E = in_sizes[2] / 2;
    const int chans[6] = {9, 16, 16, 32, 32, 64};

    const float* data[2] = {(const float*)d_in[0], (const float*)d_in[1]};
    const int*   ei[2]   = {(const int*)d_in[2],   (const int*)d_in[3]};
    auto bp = [&](int s, int b, int f) -> const float* {
        return (const float*)d_in[4 + (s * 5 + b) * 14 + f];
    };
    const float* hw1 = (const float*)d_in[144];
    const float* hb1 = (const float*)d_in[145];
    const float* hw2 = (const float*)d_in[146];
    const float* hb2 = (const float*)d_in[147];
    const float* hw3 = (const float*)d_in[148];
    const float* hb3 = (const float*)d_in[149];

    // Workspace layout (float offsets)
    float* W = (float*)d_ws;
    float* concatV = W;                 // 262144  (stream0 | stream1 final maps)
    float* ping0   = W + 262144;        // 131072
    float* ping1   = W + 393216;        // 131072
    float* res     = W + 524288;        // 131072
    float* col     = W + 655360;        // 655360  (packed B, Kpad<=320 x 2048)
    float* xl      = W + 1310720;       // 131072  [N,Ch]
    float* xr      = W + 1441792;       // 131072  [N,Ch]
    float* gat     = W + 1572864;       // 131072  [N,Ch]
    float* h1      = W + 1703936;       // 131072  [N,Ch]
    float* tc      = W + 1835008;       // 131072  [Ch,N]
    float* mbuf    = W + 1966080;       // 2048
    float* den     = W + 1968128;       // 2048
    float* ebuf    = W + 1970176;       // E + N = 67584
    float* y1      = W + 2037760;       // 1024
    float* y2      = W + 2038784;       // 512
    float* apad    = W + 2039296;       // 20480  (padded weights, <= 64x320)

    for (int s = 0; s < 2; ++s) {
        const float* x   = data[s];
        const int*   src = ei[s];
        const int*   dst = ei[s] + E;

        for (int b = 0; b < 5; ++b) {
            const int cin  = chans[b], ch = chans[b + 1];
            const int kc   = cin * 5;
            const int kcp  = (kc + 3) & ~3;
            const int cinp = (cin + 3) & ~3;
            const int ET   = E + N;

            // skip path: res = relu(conv1d_same(x, Wskip) + bskip)   [ch,N]
            pack_w<<<(ch * kcp + 255) / 256, 256, 0, stream>>>(bp(s, b, 0), apad, ch, kc, kcp);
            pack_im2col5<<<(kcp * N + 255) / 256, 256, 0, stream>>>(x, col, cin, kcp);
            wmma_gemm_f32<<<dim3(N / 64, ch / 16), 128, 0, stream>>>(
                apad, col, bp(s, b, 1), res, kcp, N, 1, 1);

            // GATv2 projections on raw-view x[n,c]=flat[n*cin+c]  ->  xl/xr [N,ch]
            pack_rawview<<<(cinp * N + 255) / 256, 256, 0, stream>>>(x, col, cin, cinp);
            pack_w<<<(ch * cinp + 255) / 256, 256, 0, stream>>>(bp(s, b, 2), apad, ch, cin, cinp);
            wmma_gemm_f32<<<dim3(N / 64, ch / 16), 128, 0, stream>>>(
                apad, col, bp(s, b, 3), xl, cinp, 1, ch, 0);
            pack_w<<<(ch * cinp + 255) / 256, 256, 0, stream>>>(bp(s, b, 4), apad, ch, cin, cinp);
            wmma_gemm_f32<<<dim3(N / 64, ch / 16), 128, 0, stream>>>(
                apad, col, bp(s, b, 5), xr, cinp, 1, ch, 0);

            // edge softmax + aggregation (E edges + N self loops)
            gat_init<<<(N * ch + 255) / 256, 256, 0, stream>>>(mbuf, den, gat, N * ch);
            edge_logits<<<(ET + 255) / 256, 256, 0, stream>>>(
                xl, xr, bp(s, b, 6), src, dst, ebuf, mbuf, E, ch);
            edge_exp<<<(ET + 255) / 256, 256, 0, stream>>>(ebuf, mbuf, den, dst, E);
            edge_aggr<<<(ET + 255) / 256, 256, 0, stream>>>(
                ebuf, den, xl, src, dst, gat, E, ch);

            // h1 = relu(BN(gat + gbias))   [N,ch]
            bn_relu_nodefirst<<<(N * ch + 255) / 256, 256, 0, stream>>>(
                gat, bp(s, b, 7), bp(s, b, 10), bp(s, b, 11), bp(s, b, 12),
                bp(s, b, 13), h1, N * ch, ch);

            // temporal conv on raw-view [ch,N] of h1:  tc = conv1d_same(h1, Wt) + bt
            const int kt  = ch * 5;
            const int ktp = (kt + 3) & ~3;
            pack_w<<<(ch * ktp + 255) / 256, 256, 0, stream>>>(bp(s, b, 8), apad, ch, kt, ktp);
            pack_im2col5<<<(ktp * N + 255) / 256, 256, 0, stream>>>(h1, col, ch, ktp);
            wmma_gemm_f32<<<dim3(N / 64, ch / 16), 128, 0, stream>>>(
                apad, col, bp(s, b, 9), tc, ktp, N, 1, 0);

            // out = res + relu(BN(tc))   [ch,N]; final block writes concat slot
            float* outb = (b == 4) ? (concatV + (size_t)s * 64 * N)
                                   : ((b & 1) ? ping1 : ping0);
            resbn_add<<<(N * ch + 255) / 256, 256, 0, stream>>>(
                tc, res, bp(s, b, 10), bp(s, b, 11), bp(s, b, 12), bp(s, b, 13),
                outb, N * ch);
            x = outb;
        }
    }

    // Head MLP: 262144 -> 1024 -> 512 -> 3 (sigmoid)
    gemv_act<<<1024, 256, 0, stream>>>(hw1, concatV, hb1, y1, 64 * N * 2, 1);
    gemv_act<<<512,  256, 0, stream>>>(hw2, y1, hb2, y2, 1024, 1);
    gemv_act<<<3,    256, 0, stream>>>(hw3, y2, hb3, (float*)d_out, 512, 2);
}